// DynamicalSystem_59880434041448
// MI455X (gfx1250) — compile-verified
//
#include <hip/hip_runtime.h>

typedef __attribute__((ext_vector_type(2))) float v2f;
typedef __attribute__((ext_vector_type(8))) float v8f;

#define DT 0.01f
#define NN 4096
#define BK 32  // K chunk (16 K-pairs)

#define WMMA_F32(A, B, C) \
    __builtin_amdgcn_wmma_f32_16x16x4_f32(false, (A), false, (B), (short)0, (C), false, false)

// ---------------------------------------------------------------------------
// Phase 1: M = I + DT*a   (64 MB elementwise)
// ---------------------------------------------------------------------------
__global__ __launch_bounds__(256) void build_M(const float* __restrict__ a,
                                               float* __restrict__ dst) {
    size_t idx = (size_t)blockIdx.x * 256 + threadIdx.x;
    int row = (int)(idx / NN);
    int col = (int)(idx % NN);
    dst[idx] = a[idx] * DT + (row == col ? 1.0f : 0.0f);
}

// ---------------------------------------------------------------------------
// Squaring GEMM: D = A * B, all 4096x4096 dense, 128x128 macro tile,
// 256 threads = 8 wave32 waves; each wave: 32x64 strip = 2x4 WMMA tiles.
// LDS holds K-pair fragments so every WMMA operand is one ds_load_b64.
// ---------------------------------------------------------------------------
__global__ __launch_bounds__(256) void gemm_sq(const float* __restrict__ A,
                                               const float* __restrict__ B,
                                               float* __restrict__ D) {
    __shared__ v2f AsP[16][129];  // [k-pair][m], +1 pad
    __shared__ v2f BsP[16][129];  // [k-pair][n], +1 pad

    const int t = threadIdx.x;
    const int lane = t & 31;
    const int wave = t >> 5;
    const int wm = wave >> 1;     // 0..3 : 32-row strip
    const int wn = wave & 1;      // 0..1 : 64-col strip
    const int lhalf = lane >> 4;  // 0|1
    const int lmod = lane & 15;

    const int row0 = blockIdx.y * 128;
    const int col0 = blockIdx.x * 128;

    // staging coords: A -> 8 v2f per thread, B -> 8 v2f per thread
    const int pa = t & 15, ma = t >> 4;    // A: pair pa, rows ma+16i
    const int nb = t & 127, pb0 = t >> 7;  // B: col nb, pairs pb0+2i

    v2f ar[8], br[8];
    auto loadA = [&](int k0) {
#pragma unroll
        for (int i = 0; i < 8; ++i)
            ar[i] = *(const v2f*)&A[(size_t)(row0 + ma + 16 * i) * NN + k0 + 2 * pa];
    };
    auto loadB = [&](int k0) {
#pragma unroll
        for (int i = 0; i < 8; ++i) {
            const int p = pb0 + 2 * i;
            v2f v;
            v.x = B[(size_t)(k0 + 2 * p) * NN + col0 + nb];
            v.y = B[(size_t)(k0 + 2 * p + 1) * NN + col0 + nb];
            br[i] = v;
        }
    };

    loadA(0);
    loadB(0);
    v8f acc[2][4] = {};

    for (int k0 = 0; k0 < NN; k0 += BK) {
#pragma unroll
        for (int i = 0; i < 8; ++i) AsP[pa][ma + 16 * i] = ar[i];
#pragma unroll
        for (int i = 0; i < 8; ++i) BsP[pb0 + 2 * i][nb] = br[i];
        __syncthreads();

        if (k0 + BK < NN) {  // prefetch next chunk while computing this one
            loadA(k0 + BK);
            loadB(k0 + BK);
        }

#pragma unroll
        for (int kq = 0; kq < BK; kq += 4) {
            const int p = (kq >> 1) + lhalf;
            v2f af0 = AsP[p][32 * wm + lmod];
            v2f af1 = AsP[p][32 * wm + 16 + lmod];
            v2f bf0 = BsP[p][64 * wn + lmod];
            v2f bf1 = BsP[p][64 * wn + 16 + lmod];
            v2f bf2 = BsP[p][64 * wn + 32 + lmod];
            v2f bf3 = BsP[p][64 * wn + 48 + lmod];
            acc[0][0] = WMMA_F32(af0, bf0, acc[0][0]);
            acc[0][1] = WMMA_F32(af0, bf1, acc[0][1]);
            acc[0][2] = WMMA_F32(af0, bf2, acc[0][2]);
            acc[0][3] = WMMA_F32(af0, bf3, acc[0][3]);
            acc[1][0] = WMMA_F32(af1, bf0, acc[1][0]);
            acc[1][1] = WMMA_F32(af1, bf1, acc[1][1]);
            acc[1][2] = WMMA_F32(af1, bf2, acc[1][2]);
            acc[1][3] = WMMA_F32(af1, bf3, acc[1][3]);
        }
        __syncthreads();
    }

#pragma unroll
    for (int r = 0; r < 2; ++r)
#pragma unroll
        for (int c = 0; c < 4; ++c) {
            const int m = row0 + 32 * wm + 16 * r + 8 * lhalf;
            const int n = col0 + 64 * wn + 16 * c + lmod;
#pragma unroll
            for (int g = 0; g < 8; ++g) D[(size_t)(m + g) * NN + n] = acc[r][c][g];
        }
}

// ---------------------------------------------------------------------------
// Rollout GEMM: D_rows = A_rows * (I + DT*a).  A/D rows gathered with stride
// rsA/rsD (64 interleaved trajectory blocks). 64x64 tile, 8 waves, each wave
// 16x32 with even/odd split accumulators.
// ---------------------------------------------------------------------------
__global__ __launch_bounds__(256) void gemm_roll(const float* __restrict__ A,
                                                 long rsA,
                                                 const float* __restrict__ B,
                                                 float* __restrict__ D,
                                                 long rsD) {
    __shared__ v2f AsP[16][65];
    __shared__ v2f BsP[16][65];

    const int t = threadIdx.x;
    const int lane = t & 31;
    const int wave = t >> 5;
    const int wm = wave >> 1;     // 0..3 : 16-row strip
    const int wn = wave & 1;      // 0..1 : 32-col strip
    const int lhalf = lane >> 4;
    const int lmod = lane & 15;

    const int col0 = blockIdx.x * 64;

    const int pa = t & 15, ma = t >> 4;   // A: 4 v2f/thread, rows ma+16i
    const int nb = t & 63, pb0 = t >> 6;  // B: 4 v2f/thread, pairs pb0+4i

    v2f ar[4], br[4];
    auto loadA = [&](int k0) {
#pragma unroll
        for (int i = 0; i < 4; ++i)
            ar[i] = *(const v2f*)&A[(size_t)(ma + 16 * i) * rsA + k0 + 2 * pa];
    };
    auto loadB = [&](int k0) {
#pragma unroll
        for (int i = 0; i < 4; ++i) {
            const int p = pb0 + 4 * i;
            v2f v;
            v.x = B[(size_t)(k0 + 2 * p) * NN + col0 + nb];
            v.y = B[(size_t)(k0 + 2 * p + 1) * NN + col0 + nb];
            br[i] = v;
        }
    };

    loadA(0);
    loadB(0);
    v8f accE0 = {}, accO0 = {}, accE1 = {}, accO1 = {};

    for (int k0 = 0; k0 < NN; k0 += BK) {
#pragma unroll
        for (int i = 0; i < 4; ++i) AsP[pa][ma + 16 * i] = ar[i];
#pragma unroll
        for (int i = 0; i < 4; ++i) {
            const int p = pb0 + 4 * i;
            v2f v = br[i];  // synthesize I + DT*a while committing to LDS
            v.x = v.x * DT + ((k0 + 2 * p) == (col0 + nb) ? 1.0f : 0.0f);
            v.y = v.y * DT + ((k0 + 2 * p + 1) == (col0 + nb) ? 1.0f : 0.0f);
            BsP[p][nb] = v;
        }
        __syncthreads();

        if (k0 + BK < NN) {
            loadA(k0 + BK);
            loadB(k0 + BK);
        }

#pragma unroll
        for (int kq = 0; kq < BK; kq += 4) {
            const int p = (kq >> 1) + lhalf;
            v2f af = AsP[p][16 * wm + lmod];
            v2f bf0 = BsP[p][32 * wn + lmod];
            v2f bf1 = BsP[p][32 * wn + 16 + lmod];
            if ((kq & 4) == 0) {
                accE0 = WMMA_F32(af, bf0, accE0);
                accE1 = WMMA_F32(af, bf1, accE1);
            } else {
                accO0 = WMMA_F32(af, bf0, accO0);
                accO1 = WMMA_F32(af, bf1, accO1);
            }
        }
        __syncthreads();
    }

    const int m = 16 * wm + 8 * lhalf;
    const int n0 = col0 + 32 * wn + lmod;
#pragma unroll
    for (int g = 0; g < 8; ++g) {
        D[(size_t)(m + g) * rsD + n0] = accE0[g] + accO0[g];
        D[(size_t)(m + g) * rsD + n0 + 16] = accE1[g] + accO1[g];
    }
}

// ---------------------------------------------------------------------------
// Seed GEMV: outS = s * B   (B = M^64)
// ---------------------------------------------------------------------------
__global__ __launch_bounds__(256) void gemv_seed(const float* __restrict__ s,
                                                 const float* __restrict__ B,
                                                 float* __restrict__ outS) {
    const int j = blockIdx.x * 256 + threadIdx.x;
    float acc = 0.f;
    for (int k = 0; k < NN; ++k) acc += s[k] * B[(size_t)k * NN + j];
    outS[j] = acc;
}

__global__ __launch_bounds__(256) void copy_vec(const float* __restrict__ src,
                                                float* __restrict__ dst) {
    const int i = blockIdx.x * 256 + threadIdx.x;
    dst[i] = src[i];
}

// ---------------------------------------------------------------------------
extern "C" void kernel_launch(void* const* d_in, const int* in_sizes, int n_in,
                              void* d_out, int out_size, void* d_ws, size_t ws_size,
                              hipStream_t stream) {
    (void)in_sizes; (void)n_in; (void)out_size; (void)ws_size;
    const float* y = (const float*)d_in[0];
    const float* a = (const float*)d_in[1];
    float* Y = (float*)d_out;           // [4096, 4096] outputs
    float* bufA = (float*)d_ws;         // 64 MB power buffer
    float* S = bufA + (size_t)NN * NN;  // [64, 4096] seeds
    float* bufB = Y;                    // reuse d_out as ping-pong scratch

    dim3 blk(256);

    // 1) M = I + DT*a  -> bufA
    build_M<<<dim3((unsigned)(((size_t)NN * NN) / 256)), blk, 0, stream>>>(a, bufA);

    // 2) six squarings: M^64 ends in bufA (even number of swaps)
    dim3 sgrid(NN / 128, NN / 128);
    const float* src = bufA;
    float* dst = bufB;
    for (int i = 0; i < 6; ++i) {
        gemm_sq<<<sgrid, blk, 0, stream>>>(src, src, dst);
        float* tmp = (float*)src;
        src = dst;
        dst = tmp;
    }
    const float* B64 = src;

    // 3) seeds: S[0] = y ; S[b] = S[b-1] * M^64
    copy_vec<<<NN / 256, blk, 0, stream>>>(y, S);
    for (int b = 1; b < 64; ++b)
        gemv_seed<<<NN / 256, blk, 0, stream>>>(S + (size_t)(b - 1) * NN, B64,
                                                S + (size_t)b * NN);

    // 4) rollout: 64 lock-step inner steps over 64 parallel blocks
    dim3 rgrid(NN / 64, 1);
    for (int j = 0; j < 64; ++j) {
        const float* Abase = (j == 0) ? (const float*)S : (const float*)(Y + (size_t)(j - 1) * NN);
        long rsA = (j == 0) ? (long)NN : (long)64 * NN;
        gemm_roll<<<rgrid, blk, 0, stream>>>(Abase, rsA, a, Y + (size_t)j * NN,
                                             (long)64 * NN);
    }
}